// Generator_67791763800697
// MI455X (gfx1250) — compile-verified
//
#include <hip/hip_runtime.h>
#include <math.h>
#include <float.h>

#define Bsz 64
#define Ssz 512
#define Hsz 400
#define Dsz 400
#define Vsz 32000
#define Lsz 16
#define G3H 1200   // 3*H

typedef float v2f __attribute__((ext_vector_type(2)));
typedef float v8f __attribute__((ext_vector_type(8)));

// ---------------------------------------------------------------- init
__global__ void zero_kernel(float* __restrict__ p, int n) {
  int i = blockIdx.x * blockDim.x + threadIdx.x;
  if (i < n) p[i] = 0.0f;
}

__global__ void init_state(const float* __restrict__ eh,       // [1,B,H]
                           const float* __restrict__ slot_emb, // [NSLOTS,D]
                           const int*   __restrict__ slot,     // scalar
                           float* __restrict__ h,               // [B,H]
                           float* __restrict__ x) {             // [B,D]
  int i = blockIdx.x * blockDim.x + threadIdx.x;
  if (i < Bsz * Hsz) h[i] = eh[i];
  if (i < Bsz * Dsz) {
    int d = i % Dsz;
    x[i] = slot_emb[(*slot) * Dsz + d];
  }
}

// ---------------------------------------------------------------- GRU gate GEMMs (WMMA f32 16x16x4)
// Computes gi = x @ W_ih^T + b_ih  and  gh = h @ W_hh^T + b_hh.
// One wave (32 threads) per 16x16 output tile. Tiles 0..299 -> gi, 300..599 -> gh.
__global__ __launch_bounds__(32)
void gemm_gates(const float* __restrict__ x,    // [B,D]
                const float* __restrict__ h,    // [B,H]
                const float* __restrict__ W_ih, // [3H,D]
                const float* __restrict__ W_hh, // [3H,H]
                const float* __restrict__ b_ih, // [3H]
                const float* __restrict__ b_hh, // [3H]
                float* __restrict__ gi,         // [B,3H]
                float* __restrict__ gh) {       // [B,3H]
  const int K = 400;          // D == H
  const int tile = blockIdx.x;          // 0..599
  const bool isGH = (tile >= 300);
  const int t  = isGH ? tile - 300 : tile;
  const int m0 = (t / 75) * 16;         // 64/16 = 4 M-tiles
  const int n0 = (t % 75) * 16;         // 1200/16 = 75 N-tiles

  const float* A    = isGH ? h    : x;
  const float* W    = isGH ? W_hh : W_ih;
  const float* bias = isGH ? b_hh : b_ih;
  float*       out  = isGH ? gh   : gi;

  const int lane = threadIdx.x & 31;
  const int half = lane >> 4;    // 0: K/K+1, 1: K+2/K+3
  const int l16  = lane & 15;
  const int koff = half * 2;

  const float* arow = A + (size_t)(m0 + l16) * K;   // A row, M = m0+l16
  const float* wrow = W + (size_t)(n0 + l16) * K;   // B[k][n] = W[n][k]

  v8f c = {};
#pragma unroll 4
  for (int k0 = 0; k0 < K; k0 += 4) {
    v2f a = *(const v2f*)(arow + k0 + koff);   // A[M][k0+koff], A[M][k0+koff+1]
    v2f b = *(const v2f*)(wrow + k0 + koff);   // B[k0+koff][N], B[k0+koff+1][N]
    c = __builtin_amdgcn_wmma_f32_16x16x4_f32(false, a, false, b, (short)0, c,
                                              false, false);
  }

  // C/D layout: VGPR r -> row m0 + r + 8*half, col n0 + l16
  const int n = n0 + l16;
  const float bn = bias[n];
#pragma unroll
  for (int r = 0; r < 8; ++r) {
    int m = m0 + r + half * 8;
    out[(size_t)m * G3H + n] = c[r] + bn;
  }
}

// ---------------------------------------------------------------- gate fuse (in-place h update)
__global__ void gru_gate(const float* __restrict__ gi,
                         const float* __restrict__ gh,
                         float* __restrict__ h) {   // [B,H] in/out
  int idx = blockIdx.x * blockDim.x + threadIdx.x;
  if (idx >= Bsz * Hsz) return;
  int b = idx / Hsz, j = idx % Hsz;
  const float* gib = gi + (size_t)b * G3H;
  const float* ghb = gh + (size_t)b * G3H;
  float r = 1.0f / (1.0f + __expf(-(gib[j]           + ghb[j])));
  float z = 1.0f / (1.0f + __expf(-(gib[Hsz + j]     + ghb[Hsz + j])));
  float n = tanhf(gib[2 * Hsz + j] + r * ghb[2 * Hsz + j]);
  h[idx] = (1.0f - z) * n + z * h[idx];
}

// ---------------------------------------------------------------- attention scores (wide grid, b128 loads)
// Grid: B*8 blocks of 256 threads. Block handles 64 rows of one batch entry.
// Wave w (of 8) handles rows [chunk*64 + w*8, +8).
__global__ __launch_bounds__(256)
void attn_scores(const float* __restrict__ h,     // [B,H] (h_new)
                 const float* __restrict__ enc,   // [B,S,H]
                 const int*   __restrict__ lens,  // [B]
                 float* __restrict__ scores) {    // [B,S]
  __shared__ __align__(16) float hv[Hsz];

  const int b     = blockIdx.x >> 3;
  const int chunk = blockIdx.x & 7;
  const int tid   = threadIdx.x;
  const int lane  = tid & 31;
  const int wave  = tid >> 5;
  const int len   = lens[b];

  for (int i = tid; i < Hsz; i += 256) hv[i] = h[(size_t)b * Hsz + i];
  __syncthreads();

  const float4* hv4 = (const float4*)hv;
  const int s_base = chunk * 64 + wave * 8;
#pragma unroll
  for (int i = 0; i < 8; ++i) {
    const int s = s_base + i;
    const float* row = enc + ((size_t)b * Ssz + s) * Hsz;
    const float4* row4 = (const float4*)row;
    float acc = 0.0f;
    // 3 x 128-float vectorized chunks (covers 384 of 400)
#pragma unroll
    for (int j = 0; j < 3; ++j) {
      float4 r4 = row4[lane + 32 * j];
      float4 h4 = hv4[lane + 32 * j];
      acc += r4.x * h4.x + r4.y * h4.y + r4.z * h4.z + r4.w * h4.w;
    }
    // 16-float tail (elements 384..399) handled by lanes 0..15
    if (lane < 16) acc += row[384 + lane] * hv[384 + lane];
    for (int off = 16; off > 0; off >>= 1) acc += __shfl_down(acc, off, 32);
    if (lane == 0) scores[(size_t)b * Ssz + s] = (s < len) ? acc : -FLT_MAX;
  }
}

// ---------------------------------------------------------------- softmax + scatter + argmax + next-x
__global__ __launch_bounds__(256)
void attn_finish(const float* __restrict__ scores, // [B,S]
                 const int*   __restrict__ lens,   // [B]
                 const int*   __restrict__ uttrs,  // [B,S]
                 const int*   __restrict__ tgt,    // [B,L]
                 const int*   __restrict__ tf,     // scalar
                 const float* __restrict__ emb,    // [V,D]
                 float* __restrict__ x,            // [B,D] (next input)
                 float* __restrict__ out_ptr,      // [B,L,V]
                 float* __restrict__ preds_out,    // [L,B]
                 int t) {
  __shared__ float sc[Ssz];
  __shared__ float rv[256];
  __shared__ int   ri[256];
  __shared__ float s_max, s_sum;
  __shared__ int   s_nxt;

  const int b   = blockIdx.x;
  const int tid = threadIdx.x;
  const int len = lens[b];

  sc[tid]       = scores[(size_t)b * Ssz + tid];
  sc[tid + 256] = scores[(size_t)b * Ssz + tid + 256];
  __syncthreads();

  // block max
  rv[tid] = fmaxf(sc[tid], sc[tid + 256]);
  __syncthreads();
  for (int off = 128; off > 0; off >>= 1) {
    if (tid < off) rv[tid] = fmaxf(rv[tid], rv[tid + off]);
    __syncthreads();
  }
  if (tid == 0) s_max = rv[0];
  __syncthreads();
  const float maxv = s_max;

  // exp (masked lanes -> 0), keep raw values for argmax
  float v0 = sc[tid], v1 = sc[tid + 256];
  float e0 = __expf(v0 - maxv), e1 = __expf(v1 - maxv);
  sc[tid] = e0; sc[tid + 256] = e1;
  float psum = e0 + e1;
  float av; int ai;
  if (v0 >= v1) { av = v0; ai = tid; } else { av = v1; ai = tid + 256; }

  // argmax (first-occurrence tie-break, like jnp.argmax)
  rv[tid] = av; ri[tid] = ai;
  __syncthreads();
  for (int off = 128; off > 0; off >>= 1) {
    if (tid < off) {
      float ov = rv[tid + off]; int oi = ri[tid + off];
      if (ov > rv[tid] || (ov == rv[tid] && oi < ri[tid])) { rv[tid] = ov; ri[tid] = oi; }
    }
    __syncthreads();
  }
  if (tid == 0) {
    int best = ri[0];
    int pred = uttrs[(size_t)b * Ssz + best];
    int nxt  = (*tf) ? tgt[(size_t)b * Lsz + t] : pred;
    preds_out[t * Bsz + b] = (float)pred;
    s_nxt = nxt;
  }

  // sum
  rv[tid] = psum;
  __syncthreads();
  for (int off = 128; off > 0; off >>= 1) {
    if (tid < off) rv[tid] += rv[tid + off];
    __syncthreads();
  }
  if (tid == 0) s_sum = rv[0];
  __syncthreads();

  const float inv = 1.0f / s_sum;
  float* orow = out_ptr + ((size_t)b * Lsz + t) * Vsz;
  int s0 = tid, s1 = tid + 256;
  if (s0 < len) atomicAdd(orow + uttrs[(size_t)b * Ssz + s0], sc[s0] * inv);
  if (s1 < len) atomicAdd(orow + uttrs[(size_t)b * Ssz + s1], sc[s1] * inv);

  const int nxt = s_nxt;
  for (int d = tid; d < Dsz; d += 256)
    x[(size_t)b * Dsz + d] = emb[(size_t)nxt * Dsz + d];
}

// ---------------------------------------------------------------- launch
extern "C" void kernel_launch(void* const* d_in, const int* in_sizes, int n_in,
                              void* d_out, int out_size, void* d_ws, size_t ws_size,
                              hipStream_t stream) {
  (void)in_sizes; (void)n_in; (void)ws_size;

  const float* eh       = (const float*)d_in[0];   // encoded_hidden [1,B,H]
  const float* enc      = (const float*)d_in[1];   // encoded_outputs [B,S,H]
  const int*   lens     = (const int*)  d_in[2];   // [B]
  const int*   uttrs    = (const int*)  d_in[3];   // [B,S]
  const int*   tgt      = (const int*)  d_in[4];   // [B,L]
  const int*   slot     = (const int*)  d_in[5];   // scalar
  const int*   tf       = (const int*)  d_in[6];   // scalar
  const float* emb      = (const float*)d_in[7];   // [V,D]
  const float* slot_emb = (const float*)d_in[8];   // [NSLOTS,D]
  const float* W_ih     = (const float*)d_in[9];   // [3H,D]
  const float* W_hh     = (const float*)d_in[10];  // [3H,H]
  const float* b_ih     = (const float*)d_in[11];  // [3H]
  const float* b_hh     = (const float*)d_in[12];  // [3H]

  float* out       = (float*)d_out;                       // [B,L,V]
  float* preds_out = out + (size_t)Bsz * Lsz * Vsz;       // [L,B]

  float* ws     = (float*)d_ws;
  float* h      = ws;                 // B*H   = 25600
  float* x      = ws + 25600;         // B*D   = 25600
  float* gi     = ws + 51200;         // B*3H  = 76800
  float* gh     = ws + 128000;        // B*3H  = 76800
  float* scores = ws + 204800;        // B*S   = 32768

  zero_kernel<<<(out_size + 255) / 256, 256, 0, stream>>>(out, out_size);
  init_state<<<(Bsz * Hsz + 255) / 256, 256, 0, stream>>>(eh, slot_emb, slot, h, x);

  for (int t = 0; t < Lsz; ++t) {
    gemm_gates<<<600, 32, 0, stream>>>(x, h, W_ih, W_hh, b_ih, b_hh, gi, gh);
    gru_gate<<<(Bsz * Hsz + 255) / 256, 256, 0, stream>>>(gi, gh, h);
    attn_scores<<<Bsz * 8, 256, 0, stream>>>(h, enc, lens, scores);
    attn_finish<<<Bsz, 256, 0, stream>>>(scores, lens, uttrs, tgt, tf, emb,
                                         x, out, preds_out, t);
  }
}